// GCN_25675314496061
// MI455X (gfx1250) — compile-verified
//
#include <hip/hip_runtime.h>

// ---------------------------------------------------------------------------
// GCN (3-layer GraphConv, norm='both') for MI455X / gfx1250, wave32 + WMMA.
//
// Pipeline (matches the DGL in_f>out_f branch structure of the reference):
//   deg_out/deg_in via atomic histograms -> norm = rsqrt(max(deg,1))
//   L1: tmp = (X*ns) @ W1  [WMMA f32 16x16x4] ; agg = scatter(tmp) ;
//       h1 = relu(agg*nd + b1)
//   L2: agg = scatter(h1*ns) ; h2 = relu((agg @ W2)*nd + b2)  [fused epilogue]
//   L3: tmp = (h2*ns) @ W3 ; out = scatter(tmp)*nd + b3
// ---------------------------------------------------------------------------

typedef float v2f __attribute__((ext_vector_type(2)));
typedef float v8f __attribute__((ext_vector_type(8)));

#define N_NODES 50000

__global__ void gcn_zero(float* __restrict__ p, int n) {
  int i = blockIdx.x * blockDim.x + threadIdx.x;
  if (i < n) p[i] = 0.0f;
}

__global__ void gcn_degree(const int* __restrict__ src, const int* __restrict__ dst,
                           float* __restrict__ deg_out, float* __restrict__ deg_in, int E) {
  int e = blockIdx.x * blockDim.x + threadIdx.x;
  if (e >= E) return;
  atomicAdd(&deg_out[src[e]], 1.0f);
  atomicAdd(&deg_in[dst[e]], 1.0f);
}

__global__ void gcn_norm(float* __restrict__ p, int n) {
  int i = blockIdx.x * blockDim.x + threadIdx.x;
  if (i < n) p[i] = __frsqrt_rn(fmaxf(p[i], 1.0f));
}

// Dense GEMM C[n_rows x F] = (pre_scale * A)[n_rows x K] @ W[K x F], with
// optional fused epilogue (row scale, bias, relu). One wave per 16x16 C tile,
// V_WMMA_F32_16X16X4_F32 over the K dimension.
__global__ void gcn_gemm_wmma(const float* __restrict__ A, const float* __restrict__ W,
                              float* __restrict__ C, int n_rows, int K, int F,
                              const float* __restrict__ pre_scale,
                              const float* __restrict__ post_scale,
                              const float* __restrict__ bias, int relu) {
  const int ftiles = F >> 4;
  const int tiles  = (n_rows >> 4) * ftiles;
  const int w = blockIdx.x * (blockDim.x >> 5) + ((int)threadIdx.x >> 5);
  if (w >= tiles) return;                       // wave-uniform: EXEC stays all-1s
  const int tileM = w / ftiles;
  const int tileN = w - tileM * ftiles;
  const int lane  = threadIdx.x & 31;
  const int half  = lane >> 4;                  // lanes 16-31 carry K+2,K+3
  const int lr    = lane & 15;

  const int m = (tileM << 4) + lr;              // A-frag row for this lane
  const int n = (tileN << 4) + lr;              // B-frag col for this lane
  const float ps = pre_scale ? pre_scale[m] : 1.0f;
  const float* Arow = A + (size_t)m * K;

  v8f acc = {};
  for (int k = 0; k < K; k += 4) {
    const int koff = k + (half << 1);
    v2f a, b;
    a.x = Arow[koff]     * ps;                  // A 16x4: v0=K(+0/+2), v1=K(+1/+3)
    a.y = Arow[koff + 1] * ps;
    b.x = W[(size_t)koff * F + n];              // B 4x16 mirrored layout
    b.y = W[(size_t)(koff + 1) * F + n];
    acc = __builtin_amdgcn_wmma_f32_16x16x4_f32(
        /*neg_a=*/false, a, /*neg_b=*/false, b,
        /*c_mod=*/(short)0, acc, /*reuse_a=*/false, /*reuse_b=*/false);
  }

  // D layout: VGPR i -> rows i (lanes 0-15) and i+8 (lanes 16-31), col = lr.
#pragma unroll
  for (int i = 0; i < 8; ++i) {
    const int row = (tileM << 4) + i + (half << 3);
    const int col = (tileN << 4) + lr;
    float v = acc[i];
    if (post_scale) v *= post_scale[row];
    if (bias)       v += bias[col];
    if (relu)       v = fmaxf(v, 0.0f);
    C[(size_t)row * F + col] = v;
  }
}

// agg[dst[e], :] += msg[src[e], :] * (scale ? scale[src[e]] : 1)
// One thread per (edge, float4-chunk); F/4 = 1<<chunk_shift.
__global__ void gcn_scatter(const float* __restrict__ msg, const int* __restrict__ src,
                            const int* __restrict__ dst, const float* __restrict__ scale,
                            float* __restrict__ agg, int n_edges, int F, int chunk_shift) {
  int idx = blockIdx.x * blockDim.x + threadIdx.x;
  int e = idx >> chunk_shift;
  if (e >= n_edges) return;
  int c = idx & ((1 << chunk_shift) - 1);
  int s = src[e], d = dst[e];
  float4 v = ((const float4*)(msg + (size_t)s * F))[c];
  if (scale) { float sc = scale[s]; v.x *= sc; v.y *= sc; v.z *= sc; v.w *= sc; }
  float* out = agg + (size_t)d * F + (c << 2);
  atomicAdd(out + 0, v.x);   // global_atomic_add_f32, no-return (STOREcnt path)
  atomicAdd(out + 1, v.y);
  atomicAdd(out + 2, v.z);
  atomicAdd(out + 3, v.w);
}

// buf[i] = maybe_relu(buf[i] * nd[row] + bias[col]),  F = 1<<f_shift
__global__ void gcn_epilogue(float* __restrict__ buf, const float* __restrict__ nd,
                             const float* __restrict__ bias, int total, int f_shift, int relu) {
  int i = blockIdx.x * blockDim.x + threadIdx.x;
  if (i >= total) return;
  int row = i >> f_shift;
  int col = i & ((1 << f_shift) - 1);
  float v = buf[i] * nd[row] + bias[col];
  if (relu) v = fmaxf(v, 0.0f);
  buf[i] = v;
}

static inline unsigned cdiv(unsigned a, unsigned b) { return (a + b - 1) / b; }

extern "C" void kernel_launch(void* const* d_in, const int* in_sizes, int n_in,
                              void* d_out, int out_size, void* d_ws, size_t ws_size,
                              hipStream_t stream) {
  const float* features = (const float*)d_in[0];   // [N, 256]
  const float* W1 = (const float*)d_in[1];         // [256, 128]
  const float* b1 = (const float*)d_in[2];         // [128]
  const float* W2 = (const float*)d_in[3];         // [128, 128]
  const float* b2 = (const float*)d_in[4];         // [128]
  const float* W3 = (const float*)d_in[5];         // [128, 64]
  const float* b3 = (const float*)d_in[6];         // [64]
  const int* src  = (const int*)d_in[7];           // [E]
  const int* dst  = (const int*)d_in[8];           // [E]
  float* out = (float*)d_out;                      // [N, 64]

  const int N  = N_NODES;
  const int E  = in_sizes[7];
  const int IN = in_sizes[0] / N;                  // 256
  const int H  = in_sizes[2];                      // 128
  const int CLS = in_sizes[6];                     // 64

  // Workspace layout (floats), all float4-aligned:
  float* ws   = (float*)d_ws;
  float* ns   = ws;                                // deg_out -> norm_src [N]
  float* nd   = ws + N;                            // deg_in  -> norm_dst [N]
  float* bufA = ws + 2 * N + 96;                   // [N, H] (offset 100096: 16B aligned)
  float* bufB = bufA + (size_t)N * H;              // [N, H]
  (void)ws_size; (void)n_in; (void)out_size;

  const int BS = 256;

  // ---- degrees -> norms (ns,nd contiguous: one zero / one norm pass) ----
  gcn_zero  <<<cdiv(2 * N, BS), BS, 0, stream>>>(ns, 2 * N);
  gcn_degree<<<cdiv(E, BS),     BS, 0, stream>>>(src, dst, ns, nd, E);
  gcn_norm  <<<cdiv(2 * N, BS), BS, 0, stream>>>(ns, 2 * N);

  // ---- Layer 1: in 256 > out 128 -> GEMM first, then aggregate ----
  {
    unsigned tiles = (N / 16) * (H / 16);
    gcn_gemm_wmma<<<cdiv(tiles, BS / 32), BS, 0, stream>>>(
        features, W1, bufA, N, IN, H, ns, nullptr, nullptr, 0);
    gcn_zero<<<cdiv(N * H, BS), BS, 0, stream>>>(bufB, N * H);
    gcn_scatter<<<cdiv((unsigned)E * (H / 4), BS), BS, 0, stream>>>(
        bufA, src, dst, nullptr, bufB, E, H, 5);
    gcn_epilogue<<<cdiv(N * H, BS), BS, 0, stream>>>(bufB, nd, b1, N * H, 7, 1);
  }

  // ---- Layer 2: in 128 == out 128 -> aggregate first, then GEMM (fused tail) ----
  {
    gcn_zero<<<cdiv(N * H, BS), BS, 0, stream>>>(bufA, N * H);
    gcn_scatter<<<cdiv((unsigned)E * (H / 4), BS), BS, 0, stream>>>(
        bufB, src, dst, ns, bufA, E, H, 5);
    unsigned tiles = (N / 16) * (H / 16);
    gcn_gemm_wmma<<<cdiv(tiles, BS / 32), BS, 0, stream>>>(
        bufA, W2, bufB, N, H, H, nullptr, nd, b2, 1);
  }

  // ---- Layer 3: in 128 > out 64 -> GEMM first, aggregate into d_out ----
  {
    unsigned tiles = (N / 16) * (CLS / 16);
    gcn_gemm_wmma<<<cdiv(tiles, BS / 32), BS, 0, stream>>>(
        bufB, W3, bufA, N, H, CLS, ns, nullptr, nullptr, 0);
    gcn_zero<<<cdiv(N * CLS, BS), BS, 0, stream>>>(out, N * CLS);
    gcn_scatter<<<cdiv((unsigned)E * (CLS / 4), BS), BS, 0, stream>>>(
        bufA, src, dst, nullptr, out, E, CLS, 4);
    gcn_epilogue<<<cdiv(N * CLS, BS), BS, 0, stream>>>(out, nd, b3, N * CLS, 6, 0);
  }
}